// QuatBlockSSM_17102559773341
// MI455X (gfx1250) — compile-verified
//
#include <hip/hip_runtime.h>
#include <hip/hip_bf16.h>
#include <math.h>
#include <stdint.h>

typedef __attribute__((ext_vector_type(2))) float        v2f;
typedef __attribute__((ext_vector_type(8))) float        v8f;
typedef __attribute__((ext_vector_type(4))) unsigned int v4u;
typedef __attribute__((ext_vector_type(4))) int          v4i;
typedef __attribute__((ext_vector_type(8))) int          v8i;

#define D_MODEL 1536
#define BATCH   8192
#define EPS     1e-8f

#define M_TILE  32
#define N_TILE  48        // lcm(16,3): WMMA-tileable and quaternion-triple aligned
#define KC      32        // K chunk staged through LDS (32 DWORDs per tile row)
#define LDK     36        // LDS row stride in floats = 32 data + 4 pad (TDM pad unit)
#define EPAD    50        // padded stride for epilogue tiles
#define NWAVES  6
#define WGSIZE  (NWAVES * 32)

// ---------------------------------------------------------------------------
// Kernel 1: decay[m] = sigmoid(dot(x[m], W_dec) + b_dec)   (one wave per row)
// ---------------------------------------------------------------------------
__global__ __launch_bounds__(256) void decay_kernel(
    const float* __restrict__ x, const float* __restrict__ Wd,
    const float* __restrict__ bd, float* __restrict__ decay)
{
    const int wave = threadIdx.x >> 5;
    const int lane = threadIdx.x & 31;
    const int row  = blockIdx.x * 8 + wave;
    if (row >= BATCH) return;

    const float* xr = x + (size_t)row * D_MODEL;
    float acc = 0.0f;
    for (int k = lane * 4; k < D_MODEL; k += 32 * 4) {
        float4 xv = *(const float4*)(xr + k);
        float4 wv = *(const float4*)(Wd + k);
        acc += xv.x * wv.x + xv.y * wv.y + xv.z * wv.z + xv.w * wv.w;
    }
    for (int off = 16; off > 0; off >>= 1)
        acc += __shfl_xor(acc, off, 32);
    if (lane == 0) {
        float z = acc + bd[0];
        decay[row] = 1.0f / (1.0f + expf(-z));
    }
}

// ---------------------------------------------------------------------------
// TDM tile load: DMA an (nrows x KC) f32 tile (row stride D_MODEL in memory)
// into LDS at byte offset lds_off, padding each 32-DWORD row by 4 DWORDs so
// the LDS layout has the conflict-free LDK=36 stride.  Descriptor per
// CDNA5 ISA sections 8.3/8.4.  Must be executed by exactly one wave (TDM
// ignores EXEC); completion tracked by TENSORcnt of the issuing wave.
// ---------------------------------------------------------------------------
__device__ __forceinline__ void tdm_load_tile(unsigned int lds_off,
                                              const float* gsrc, int nrows)
{
    unsigned long long ga = (unsigned long long)(uintptr_t)gsrc;
    v4u g0;
    g0.x = 1u;                                   // count=1 (valid user descriptor)
    g0.y = lds_off;                              // lds_addr (bytes)
    g0.z = (unsigned int)ga;                     // global_addr[31:0]
    g0.w = ((unsigned int)(ga >> 32) & 0x01FFFFFFu) | (2u << 30); // addr[56:32] | type=2

    v8i g1;
    g1[0] = (2 << 16)        // data_size = 2 -> 4-byte elements
          | (1 << 20)        // pad_enable
          | (4 << 22)        // pad_interval code 4 -> every 32 DWORDs
          | (3 << 25);       // pad_amount  code 3 -> 4 DWORDs
    g1[1] = (KC << 16);      // tensor_dim0 = KC      (bits 79:48, low half)
    g1[2] = (nrows << 16);   // tensor_dim1 = nrows   (bits 111:80, low half)
    g1[3] = (KC << 16);      // tile_dim0   = KC      (bits 127:112)
    g1[4] = nrows;           // tile_dim1   = nrows   (bits 143:128)
    g1[5] = D_MODEL;         // tensor_dim0_stride = 1536 elements (bits 207:160)
    g1[6] = 0;
    g1[7] = 0;

    v4i z4 = {0, 0, 0, 0};
#if __has_include(<hip/amd_detail/amd_gfx1250_TDM.h>)
    v8i z8 = {0, 0, 0, 0, 0, 0, 0, 0};
    __builtin_amdgcn_tensor_load_to_lds(g0, g1, z4, z4, z8, 0);
#else
    __builtin_amdgcn_tensor_load_to_lds(g0, g1, z4, z4, 0);
#endif
}

// ---------------------------------------------------------------------------
// Kernel 2: fused dual f32-WMMA GEMM (bivector + injection) with TDM-staged,
// double-buffered LDS tiles + quaternion rotation epilogue.
// One workgroup = 32x48 output tile, 6 waves, each wave owns one 16x16
// position and both accumulators (shared A fragment).
// ---------------------------------------------------------------------------
__global__ __launch_bounds__(WGSIZE) void fused_kernel(
    const float* __restrict__ x,     const float* __restrict__ h_prev,
    const float* __restrict__ W_biv, const float* __restrict__ b_biv,
    const float* __restrict__ W_in,  const float* __restrict__ b_in,
    const float* __restrict__ decay, float* __restrict__ out)
{
    __shared__ float sx [2][M_TILE * LDK];   // x tile      32 x KC (double-buffered)
    __shared__ float swb[2][N_TILE * LDK];   // W_biv tile  48 x KC
    __shared__ float swi[2][N_TILE * LDK];   // W_in  tile  48 x KC
    __shared__ float sbiv[M_TILE * EPAD];    // biv tile    32 x 48
    __shared__ float sinj[M_TILE * EPAD];    // inj tile    32 x 48

    const int tid   = threadIdx.x;
    const int wave  = tid >> 5;           // 0..5
    const int lane  = tid & 31;
    const int l16   = lane & 15;
    const int lhalf = lane >> 4;          // 0 or 1
    const int koff  = lhalf * 2;          // K sub-offset per ISA A/B layout
    const int mi    = wave / 3;           // 0..1  (16-row sub-tile)
    const int ni    = wave % 3;           // 0..2  (16-col sub-tile)

    const int m0 = blockIdx.y * M_TILE;
    const int n0 = blockIdx.x * N_TILE;

    const float* xg  = x     + (size_t)m0 * D_MODEL;   // tile row base, col 0
    const float* wbg = W_biv + (size_t)n0 * D_MODEL;
    const float* wig = W_in  + (size_t)n0 * D_MODEL;

    const unsigned int sx_off[2]  = { (unsigned int)(uintptr_t)&sx[0][0],
                                      (unsigned int)(uintptr_t)&sx[1][0] };
    const unsigned int swb_off[2] = { (unsigned int)(uintptr_t)&swb[0][0],
                                      (unsigned int)(uintptr_t)&swb[1][0] };
    const unsigned int swi_off[2] = { (unsigned int)(uintptr_t)&swi[0][0],
                                      (unsigned int)(uintptr_t)&swi[1][0] };

    v8f c1 = {};   // bivector accumulator
    v8f c2 = {};   // injection accumulator

    // ---- prologue: stage K-chunk 0 into buffer 0 via TDM (wave 0 only) ----
    if (wave == 0) {
        tdm_load_tile(sx_off[0],  xg,  M_TILE);
        tdm_load_tile(swb_off[0], wbg, N_TILE);
        tdm_load_tile(swi_off[0], wig, N_TILE);
    }
    __builtin_amdgcn_s_wait_tensorcnt(0);   // no-op for waves with TENSORcnt==0
    __syncthreads();

    for (int k0 = 0; k0 < D_MODEL; k0 += KC) {
        const int cur = (k0 / KC) & 1;
        const int nxt = cur ^ 1;

        // ---- prefetch next K-chunk into the other buffer (overlapped) ----
        // Safe: buffer `nxt` was last read in iteration k0-KC, whose trailing
        // barrier ordered those reads before this TDM issue.
        if (wave == 0 && (k0 + KC) < D_MODEL) {
            tdm_load_tile(sx_off[nxt],  xg  + k0 + KC, M_TILE);
            tdm_load_tile(swb_off[nxt], wbg + k0 + KC, N_TILE);
            tdm_load_tile(swi_off[nxt], wig + k0 + KC, N_TILE);
        }

        // ---- WMMA over current chunk: KC/4 = 8 steps, 2 GEMMs each ----
        const float* sxc = &sx [cur][0];
        const float* sbc = &swb[cur][0];
        const float* sic = &swi[cur][0];
        #pragma unroll
        for (int kk = 0; kk < KC; kk += 4) {
            v2f a, bb1, bb2;
            const float* arow = sxc + (mi * 16 + l16) * LDK + kk + koff;
            const float* brow = sbc + (ni * 16 + l16) * LDK + kk + koff;
            const float* irow = sic + (ni * 16 + l16) * LDK + kk + koff;
            a.x   = arow[0]; a.y   = arow[1];    // A[m][k], A[m][k+1]
            bb1.x = brow[0]; bb1.y = brow[1];    // B[k][n] = W_biv[n][k]
            bb2.x = irow[0]; bb2.y = irow[1];    // B[k][n] = W_in [n][k]
            c1 = __builtin_amdgcn_wmma_f32_16x16x4_f32(
                     false, a, false, bb1, (short)0, c1, false, false);
            c2 = __builtin_amdgcn_wmma_f32_16x16x4_f32(
                     false, a, false, bb2, (short)0, c2, false, false);
        }

        // wave 0 waits for its prefetch DMA to land, then everyone syncs:
        // next iteration may read buffer `nxt` and overwrite buffer `cur`.
        __builtin_amdgcn_s_wait_tensorcnt(0);
        __syncthreads();
    }

    // ---- bias add + dump tiles to LDS for the triple-aligned epilogue ----
    {
        const int n_local = ni * 16 + l16;
        const float bbias = b_biv[n0 + n_local];
        const float ibias = b_in [n0 + n_local];
        #pragma unroll
        for (int r = 0; r < 8; ++r) {
            int m_local = mi * 16 + r + 8 * lhalf;   // C layout: VGPR r -> M = r + 8*half
            sbiv[m_local * EPAD + n_local] = c1[r] + bbias;
            sinj[m_local * EPAD + n_local] = c2[r] + ibias;
        }
    }
    __syncthreads();

    // ---- quaternion-rotate epilogue: 32 rows x 16 triples = 512 tasks ----
    for (int t = tid; t < M_TILE * (N_TILE / 3); t += WGSIZE) {
        const int row  = t / (N_TILE / 3);
        const int trip = t % (N_TILE / 3);
        const int m    = m0 + row;
        const int nb   = trip * 3;

        const float bx = sbiv[row * EPAD + nb + 0];
        const float by = sbiv[row * EPAD + nb + 1];
        const float bz = sbiv[row * EPAD + nb + 2];

        const size_t gofs = (size_t)m * D_MODEL + n0 + nb;
        const float vx = h_prev[gofs + 0];
        const float vy = h_prev[gofs + 1];
        const float vz = h_prev[gofs + 2];

        float nrm = sqrtf(bx * bx + by * by + bz * bz);
        nrm = fmaxf(nrm, EPS);
        const float ha = 0.5f * nrm;
        const float w  = cosf(ha);
        const float s  = sinf(ha) / nrm;
        const float qx =  s * bz;
        const float qy = -s * by;
        const float qz =  s * bx;

        const float tx = 2.0f * (qy * vz - qz * vy);
        const float ty = 2.0f * (qz * vx - qx * vz);
        const float tz = 2.0f * (qx * vy - qy * vx);

        const float rx = vx + w * tx + (qy * tz - qz * ty);
        const float ry = vy + w * ty + (qz * tx - qx * tz);
        const float rz = vz + w * tz + (qx * ty - qy * tx);

        const float d  = decay[m];
        out[gofs + 0] = d * rx + sinj[row * EPAD + nb + 0];
        out[gofs + 1] = d * ry + sinj[row * EPAD + nb + 1];
        out[gofs + 2] = d * rz + sinj[row * EPAD + nb + 2];
    }
}

// ---------------------------------------------------------------------------
extern "C" void kernel_launch(void* const* d_in, const int* in_sizes, int n_in,
                              void* d_out, int out_size, void* d_ws, size_t ws_size,
                              hipStream_t stream) {
    (void)in_sizes; (void)n_in; (void)out_size; (void)ws_size;
    const float* x      = (const float*)d_in[0];
    const float* h_prev = (const float*)d_in[1];
    const float* W_biv  = (const float*)d_in[2];
    const float* b_biv  = (const float*)d_in[3];
    const float* W_dec  = (const float*)d_in[4];
    const float* b_dec  = (const float*)d_in[5];
    const float* W_in   = (const float*)d_in[6];
    const float* b_in   = (const float*)d_in[7];
    float* out   = (float*)d_out;
    float* decay = (float*)d_ws;   // 8192 floats = 32 KB of workspace

    decay_kernel<<<BATCH / 8, 256, 0, stream>>>(x, W_dec, b_dec, decay);

    dim3 grid(D_MODEL / N_TILE, BATCH / M_TILE);   // (32, 256)
    fused_kernel<<<grid, WGSIZE, 0, stream>>>(x, h_prev, W_biv, b_biv,
                                              W_in, b_in, decay, out);
}